// RetinaHeadLoss_14396730376698
// MI455X (gfx1250) — compile-verified
//
#include <hip/hip_runtime.h>
#include <hip/hip_bf16.h>

typedef __attribute__((ext_vector_type(2))) float v2f;
typedef __attribute__((ext_vector_type(8))) float v8f;

#define F_ALPHA 0.25f
#define F_EPS   1e-4f
#define MAXT    128

// ---------------------------------------------------------------------------
// Wave-level sum using the CDNA5 matrix pipe:
// A (16x4 f32): lane L<16 -> A[L][0]=s (VGPR0), A[L][1]=0 (VGPR1);
//               lane L>=16 -> A[L-16][2]=s, A[L-16][3]=0.
// B (4x16) = all ones  =>  D[m][n] = s_m + s_{m+16} for every column n.
// Lanes 0-15 hold rows 0..7 in d[0..7]; lanes 16-31 hold rows 8..15.
// One shfl_xor(16) completes the 32-lane total (all lanes get it).
// ---------------------------------------------------------------------------
__device__ __forceinline__ float wave_sum_wmma(float s) {
    v2f a; a[0] = s;    a[1] = 0.0f;
    v2f b; b[0] = 1.0f; b[1] = 1.0f;
    v8f c = {};
    c = __builtin_amdgcn_wmma_f32_16x16x4_f32(false, a, false, b,
                                              (short)0, c, false, false);
    float t = c[0] + c[1] + c[2] + c[3] + c[4] + c[5] + c[6] + c[7];
    t += __shfl_xor(t, 16, 32);
    return t;
}

// Block-wide sum (all threads must call; returns total to every thread).
__device__ __forceinline__ float block_sum(float v) {
    __shared__ float sh[32];
    float w = wave_sum_wmma(v);
    const int lane = threadIdx.x & 31;
    const int wid  = threadIdx.x >> 5;
    __syncthreads();
    if (lane == 0) sh[wid] = w;
    __syncthreads();
    float tot = 0.0f;
    const int nw = (blockDim.x + 31) >> 5;
    for (int i = 0; i < nw; ++i) tot += sh[i];
    return tot;
}

__device__ __forceinline__ float smooth_l1(float d) {
    d = fabsf(d);
    return (d <= (1.0f / 9.0f)) ? (4.5f * d * d) : (d - (0.5f / 9.0f));
}

// Focal term with exactly one log per element.
// state s: -1 ignore, -2 background, >=0 positive (assigned class id).
__device__ __forceinline__ float focal_term(float p, int s, int cls) {
    p = fminf(fmaxf(p, F_EPS), 1.0f - F_EPS);
    const bool one = (s >= 0) && (cls == s);
    const float q  = one ? p : (1.0f - p);        // log argument
    const float f  = 1.0f - q;                     // focal factor base
    const float w  = one ? F_ALPHA : (1.0f - F_ALPHA);
    const float loss = -w * f * f * __logf(q);
    return (s == -1) ? 0.0f : loss;
}

// ---------------------------------------------------------------------------
// Kernel 1: anchor<->target matching + smooth-L1 regression partials.
// grid = ((A+255)/256, B), block = 256
// ---------------------------------------------------------------------------
__global__ void retina_match_kernel(const float* __restrict__ anchors,   // [A,4]
                                    const float* __restrict__ targets,   // [B,T,5]
                                    const float* __restrict__ regs,      // [B,A,4]
                                    int* __restrict__ astate,            // [B,A]
                                    float* __restrict__ reg_part,        // [B,GX1]
                                    float* __restrict__ np_part,         // [B,GX1]
                                    int A, int T) {
    __shared__ float tg[MAXT * 5];
    const int b  = blockIdx.y;
    const int tt = (T < MAXT) ? T : MAXT;

    const float* tsrc = targets + (size_t)b * T * 5;
    for (int i = threadIdx.x; i < tt * 5; i += blockDim.x) tg[i] = tsrc[i];
    __syncthreads();

    const int a = blockIdx.x * blockDim.x + threadIdx.x;
    float my_reg = 0.0f;
    float my_np  = 0.0f;

    if (a < A) {
        const float ax1 = anchors[a * 4 + 0];
        const float ay1 = anchors[a * 4 + 1];
        const float ax2 = anchors[a * 4 + 2];
        const float ay2 = anchors[a * 4 + 3];
        const float areaA = (ax2 - ax1) * (ay2 - ay1);

        float best = -1.0f;
        int   bi   = 0;
        bool  any_valid = false;
        for (int t = 0; t < tt; ++t) {
            const float lab = tg[t * 5 + 4];
            if (lab == -1.0f) continue;           // padded target
            any_valid = true;
            const float tx1 = tg[t * 5 + 0];
            const float ty1 = tg[t * 5 + 1];
            const float tx2 = tg[t * 5 + 2];
            const float ty2 = tg[t * 5 + 3];
            const float iw = fmaxf(fminf(ax2, tx2) - fmaxf(ax1, tx1), 0.0f);
            const float ih = fmaxf(fminf(ay2, ty2) - fmaxf(ay1, ty1), 0.0f);
            const float inter = iw * ih;
            const float areaB = (tx2 - tx1) * (ty2 - ty1);
            const float iou = inter / (areaA + areaB - inter);
            if (iou > best) { best = iou; bi = t; }
        }

        int state;
        const bool pos = any_valid && (best >= 0.5f);
        if (!any_valid)          state = -1;       // has_valid==False -> loss 0
        else if (pos)            state = (int)tg[bi * 5 + 4];
        else if (best < 0.4f)    state = -2;       // background
        else                     state = -1;       // ignored band [0.4, 0.5)
        astate[(size_t)b * A + a] = state;

        if (pos) {
            const float mx1 = tg[bi * 5 + 0];
            const float my1 = tg[bi * 5 + 1];
            const float mx2 = tg[bi * 5 + 2];
            const float my2 = tg[bi * 5 + 3];
            const float cx = 0.5f * (ax1 + ax2), cy = 0.5f * (ay1 + ay2);
            const float w  = ax2 - ax1,          h  = ay2 - ay1;
            const float gcx = (0.5f * (mx1 + mx2) - cx) / (0.1f * w);
            const float gcy = (0.5f * (my1 + my2) - cy) / (0.1f * h);
            const float gw  = __logf((mx2 - mx1) / w) * 5.0f;   // /0.2
            const float gh  = __logf((my2 - my1) / h) * 5.0f;
            const float* r = regs + ((size_t)b * A + a) * 4;
            my_reg = smooth_l1(gcx - r[0]) + smooth_l1(gcy - r[1]) +
                     smooth_l1(gw  - r[2]) + smooth_l1(gh  - r[3]);
            my_np = 1.0f;
        }
    }

    const float reg_tot = block_sum(my_reg);
    const float np_tot  = block_sum(my_np);
    if (threadIdx.x == 0) {
        reg_part[(size_t)b * gridDim.x + blockIdx.x] = reg_tot;
        np_part [(size_t)b * gridDim.x + blockIdx.x] = np_tot;
    }
}

// ---------------------------------------------------------------------------
// Kernel 2: focal classification loss. Streams cla as float4 (C % 4 == 0 so
// each float4 lies in one anchor). grid = (GX2, B), block = 256.
// ---------------------------------------------------------------------------
__global__ void retina_focal_kernel(const float* __restrict__ clas,   // [B,A,C]
                                    const int* __restrict__ astate,   // [B,A]
                                    float* __restrict__ cls_part,     // [B,GX2]
                                    int A, int C) {
    const int b = blockIdx.y;
    const unsigned AC  = (unsigned)A * (unsigned)C;
    const unsigned AC4 = AC >> 2;
    const float4* p4 = (const float4*)(clas + (size_t)b * AC);
    const int* st = astate + (size_t)b * A;

    float local = 0.0f;
    const unsigned stride = gridDim.x * blockDim.x;
    for (unsigned i = blockIdx.x * blockDim.x + threadIdx.x; i < AC4; i += stride) {
        const float4 v = p4[i];
        const unsigned e   = i << 2;
        const unsigned anc = e / (unsigned)C;
        const int      c0  = (int)(e - anc * (unsigned)C);
        const int      s   = st[anc];
        local += focal_term(v.x, s, c0)
               + focal_term(v.y, s, c0 + 1)
               + focal_term(v.z, s, c0 + 2)
               + focal_term(v.w, s, c0 + 3);
    }
    // Tail (only if A*C not divisible by 4).
    const unsigned rem = AC - (AC4 << 2);
    if (blockIdx.x == 0 && threadIdx.x < rem) {
        const unsigned e = (AC4 << 2) + threadIdx.x;
        const unsigned anc = e / (unsigned)C;
        const int c0 = (int)(e - anc * (unsigned)C);
        local += focal_term(clas[(size_t)b * AC + e], st[anc], c0);
    }

    const float tot = block_sum(local);
    if (threadIdx.x == 0)
        cls_part[(size_t)b * gridDim.x + blockIdx.x] = tot;
}

// ---------------------------------------------------------------------------
// Kernel 3: per-image normalization. grid = (B), block = 256.
// ---------------------------------------------------------------------------
__global__ void retina_reduce_kernel(const float* __restrict__ cls_part,
                                     const float* __restrict__ reg_part,
                                     const float* __restrict__ np_part,
                                     int gx2, int gx1,
                                     float* __restrict__ per_img) {     // [B,2]
    const int b = blockIdx.x;
    float c = 0.0f, r = 0.0f, n = 0.0f;
    for (int i = threadIdx.x; i < gx2; i += blockDim.x)
        c += cls_part[(size_t)b * gx2 + i];
    for (int i = threadIdx.x; i < gx1; i += blockDim.x) {
        r += reg_part[(size_t)b * gx1 + i];
        n += np_part [(size_t)b * gx1 + i];
    }
    c = block_sum(c);
    r = block_sum(r);
    n = block_sum(n);
    if (threadIdx.x == 0) {
        per_img[b * 2 + 0] = c / fmaxf(n, 1.0f);
        per_img[b * 2 + 1] = (n > 0.0f) ? (r / (n * 4.0f)) : 0.0f;
    }
}

// ---------------------------------------------------------------------------
// Kernel 4: mean over batch -> out[0], out[1].
// ---------------------------------------------------------------------------
__global__ void retina_final_kernel(const float* __restrict__ per_img,
                                    int B, float* __restrict__ out) {
    if (threadIdx.x == 0 && blockIdx.x == 0) {
        float cm = 0.0f, rm = 0.0f;
        for (int b = 0; b < B; ++b) {
            cm += per_img[b * 2 + 0];
            rm += per_img[b * 2 + 1];
        }
        out[0] = cm / (float)B;
        out[1] = rm / (float)B;
    }
}

// ---------------------------------------------------------------------------
extern "C" void kernel_launch(void* const* d_in, const int* in_sizes, int n_in,
                              void* d_out, int out_size, void* d_ws, size_t ws_size,
                              hipStream_t stream) {
    const float* clas    = (const float*)d_in[0];   // [B,A,C]
    const float* regs    = (const float*)d_in[1];   // [B,A,4]
    const float* anchors = (const float*)d_in[2];   // [1,A,4]
    const float* targets = (const float*)d_in[3];   // [B,T,5]

    const int A = in_sizes[2] / 4;
    const int B = in_sizes[1] / (A * 4);
    const int C = in_sizes[0] / (A * B);
    const int T = in_sizes[3] / (B * 5);

    const int BLK = 256;
    const int GX1 = (A + BLK - 1) / BLK;   // match blocks per image
    const int GX2 = 512;                   // focal blocks per image

    // Workspace layout
    char* ws = (char*)d_ws;
    int*   astate   = (int*)ws;                    ws += (size_t)B * A * sizeof(int);
    float* cls_part = (float*)ws;                  ws += (size_t)B * GX2 * sizeof(float);
    float* reg_part = (float*)ws;                  ws += (size_t)B * GX1 * sizeof(float);
    float* np_part  = (float*)ws;                  ws += (size_t)B * GX1 * sizeof(float);
    float* per_img  = (float*)ws;                  ws += (size_t)B * 2 * sizeof(float);

    retina_match_kernel<<<dim3(GX1, B), BLK, 0, stream>>>(
        anchors, targets, regs, astate, reg_part, np_part, A, T);

    retina_focal_kernel<<<dim3(GX2, B), BLK, 0, stream>>>(
        clas, astate, cls_part, A, C);

    retina_reduce_kernel<<<dim3(B), BLK, 0, stream>>>(
        cls_part, reg_part, np_part, GX2, GX1, per_img);

    retina_final_kernel<<<dim3(1), 32, 0, stream>>>(
        per_img, B, (float*)d_out);
}